// SLOTDecomposer_47175920779637
// MI455X (gfx1250) — compile-verified
//
#include <hip/hip_runtime.h>
#include <math.h>

typedef float v2f __attribute__((ext_vector_type(2)));
typedef float v8f __attribute__((ext_vector_type(8)));

#define NCH   256
#define KDIM  24
#define C1K   1.191042e-08f
#define C2K   1.4387769f

// ---------------------------------------------------------------------------
// Kernel 1: fused emissivity-GEMM (f32 WMMA) + Planck + s_model + reductions.
// Block = 512 thr = 16 waves. Block covers 16 pixels x 256 channels.
// Wave w computes the 16x16 tile at channels [16w, 16w+16) via
// V_WMMA_F32_16X16X4_F32, accumulating over K=24 in 6 steps.
// ---------------------------------------------------------------------------
__global__ __launch_bounds__(512) void slot_main_kernel(
    const float* __restrict__ s_obs,
    const float* __restrict__ temperature,
    const float* __restrict__ beta,
    const float* __restrict__ view_factor,
    const float* __restrict__ s_sky,
    const float* __restrict__ s_ground,
    const float* __restrict__ phi,          // [NCH][KDIM] row-major
    const float* __restrict__ wavenumber,   // [NCH]
    float* __restrict__ s_model_out,        // d_out + 1
    float* __restrict__ ws_partials)        // 2 floats per block
{
    __shared__ float sBeta[16 * KDIM];
    __shared__ float sT[16];
    __shared__ float sVF[16];
    __shared__ float sRedA[512];
    __shared__ float sRedB[512];

    const int t       = threadIdx.x;
    const int pixBase = blockIdx.x * 16;

    // Stage the 16xK beta tile + per-pixel scalars into LDS.
    if (t < 16 * KDIM) sBeta[t] = beta[(size_t)pixBase * KDIM + t];
    if (t < 16) {
        sT[t]  = temperature[pixBase + t];
        sVF[t] = view_factor[pixBase + t];
    }
    __syncthreads();

    const int wave = t >> 5;        // 0..15 -> channel tile
    const int lane = t & 31;
    const int lm   = lane & 15;     // A: M index / B: N index
    const int hs   = lane >> 4;     // half-wave selector (K offset +2)
    const int ch   = wave * 16 + lm;

    // Accumulate emissivity tile: D = sum_k A(16x4) x B(4x16)
    v8f c = {0.f, 0.f, 0.f, 0.f, 0.f, 0.f, 0.f, 0.f};
#pragma unroll
    for (int k0 = 0; k0 < KDIM; k0 += 4) {
        v2f a, b;
        // A 16x4 layout: lane%16 = M; VGPR0/1 = K+{0,1}; lanes>=16 add K+2.
        a.x = sBeta[lm * KDIM + k0 + 2 * hs];
        a.y = sBeta[lm * KDIM + k0 + 2 * hs + 1];
        // B 4x16 layout (B[k][n] = phi[n][k]): lane%16 = N; same K striping.
        b.x = phi[ch * KDIM + k0 + 2 * hs];
        b.y = phi[ch * KDIM + k0 + 2 * hs + 1];
        c = __builtin_amdgcn_wmma_f32_16x16x4_f32(
                /*neg_a=*/false, a, /*neg_b=*/false, b,
                /*c_mod=*/(short)0, c, /*reuse_a=*/false, /*reuse_b=*/false);
    }

    // Channel-dependent constants (fixed per lane across all 8 accum rows).
    const float v    = wavenumber[ch];
    const float sky  = s_sky[ch];
    const float gnd  = s_ground[ch];
    const float c1v3 = C1K * v * v * v;
    const float c2v  = C2K * v;

    float res = 0.f, bv = 0.f;
#pragma unroll
    for (int r = 0; r < 8; ++r) {
        const int   M  = r + 8 * hs;          // C/D layout: VGPR r -> M=r (+8 hi half)
        const int   p  = pixBase + M;
        const float e  = c[r];
        const float T  = sT[M];
        const float vf = sVF[M];
        const float planck = c1v3 / expm1f(c2v / T);
        const float xamb   = vf * sky + (1.f - vf) * gnd;
        const float sm     = e * planck + (1.f - e) * xamb;
        const float obs    = s_obs[(size_t)p * NCH + ch];
        const float d      = obs - sm;
        res += d * d;
        bv  += fmaxf(0.01f - e, 0.f) + fmaxf(e - 0.99f, 0.f);
        s_model_out[(size_t)p * NCH + ch] = sm;
    }

    // Deterministic block tree-reduction of (residual, bound_violation).
    sRedA[t] = res;
    sRedB[t] = bv;
    __syncthreads();
    for (int s = 256; s > 0; s >>= 1) {
        if (t < s) { sRedA[t] += sRedA[t + s]; sRedB[t] += sRedB[t + s]; }
        __syncthreads();
    }
    if (t == 0) {
        ws_partials[2 * blockIdx.x]     = sRedA[0];
        ws_partials[2 * blockIdx.x + 1] = sRedB[0];
    }
}

// ---------------------------------------------------------------------------
// Kernel 2: smoothness  sum_j (beta[j] - 2 beta[j+1] + beta[j+2])^2
// (d_beta is analytically the clamped second-difference operator.)
// ---------------------------------------------------------------------------
__global__ __launch_bounds__(256) void slot_smooth_kernel(
    const float* __restrict__ beta, float* __restrict__ ws_out)
{
    __shared__ float sRed[256];
    const int    t  = threadIdx.x;
    const size_t p  = (size_t)blockIdx.x * 256 + t;
    const float* bp = beta + p * KDIM;

    float b[KDIM];
#pragma unroll
    for (int i = 0; i < KDIM; i += 4) {          // 96B rows are 16B aligned
        float4 q = *(const float4*)(bp + i);
        b[i] = q.x; b[i + 1] = q.y; b[i + 2] = q.z; b[i + 3] = q.w;
    }
    float s = 0.f;
#pragma unroll
    for (int j = 0; j < KDIM - 2; ++j) {
        const float d = b[j] - 2.f * b[j + 1] + b[j + 2];
        s += d * d;
    }
    sRed[t] = s;
    __syncthreads();
    for (int k = 128; k > 0; k >>= 1) {
        if (t < k) sRed[t] += sRed[t + k];
        __syncthreads();
    }
    if (t == 0) ws_out[blockIdx.x] = sRed[0];
}

// ---------------------------------------------------------------------------
// Kernel 3: final scalar combine -> d_out[0]
// ---------------------------------------------------------------------------
__global__ __launch_bounds__(256) void slot_final_kernel(
    const float* __restrict__ ws, int nBlocks1, int nBlocks2, float invNPix,
    float* __restrict__ out)
{
    __shared__ float sr[256], sb[256], ss[256];
    const int t = threadIdx.x;
    float r = 0.f, b = 0.f, s = 0.f;
    for (int i = t; i < nBlocks1; i += 256) { r += ws[2 * i]; b += ws[2 * i + 1]; }
    const float* w2 = ws + 2 * (size_t)nBlocks1;
    for (int i = t; i < nBlocks2; i += 256) s += w2[i];
    sr[t] = r; sb[t] = b; ss[t] = s;
    __syncthreads();
    for (int k = 128; k > 0; k >>= 1) {
        if (t < k) { sr[t] += sr[t + k]; sb[t] += sb[t + k]; ss[t] += ss[t + k]; }
        __syncthreads();
    }
    if (t == 0) {
        // residual + REG_LAMBDA/2 * smooth + CONSTRAINT_MU * bound, each "mean
        // over pixels of per-pixel sums" == global_sum / N_PIXELS.
        out[0] = sr[0] * invNPix + 0.5f * ss[0] * invNPix + 10.0f * sb[0] * invNPix;
    }
}

// ---------------------------------------------------------------------------
extern "C" void kernel_launch(void* const* d_in, const int* in_sizes, int n_in,
                              void* d_out, int out_size, void* d_ws, size_t ws_size,
                              hipStream_t stream)
{
    const float* s_obs       = (const float*)d_in[0];
    const float* temperature = (const float*)d_in[1];
    const float* beta        = (const float*)d_in[2];
    const float* view_factor = (const float*)d_in[3];
    const float* s_sky       = (const float*)d_in[4];
    const float* s_ground    = (const float*)d_in[5];
    const float* phi         = (const float*)d_in[6];
    // d_in[7] = d_beta: clamped second-difference operator, applied analytically.
    const float* wavenumber  = (const float*)d_in[8];

    const int nPix = in_sizes[1];       // 262144
    const int nb1  = nPix / 16;         // 16384 main blocks
    const int nb2  = nPix / 256;        // 1024 smoothness blocks

    float* out = (float*)d_out;         // out[0] = total; out+1 = s_model
    float* ws  = (float*)d_ws;          // [2*nb1] res/bv partials, then [nb2]

    slot_main_kernel<<<nb1, 512, 0, stream>>>(
        s_obs, temperature, beta, view_factor, s_sky, s_ground, phi,
        wavenumber, out + 1, ws);
    slot_smooth_kernel<<<nb2, 256, 0, stream>>>(beta, ws + 2 * (size_t)nb1);
    slot_final_kernel<<<1, 256, 0, stream>>>(ws, nb1, nb2, 1.0f / (float)nPix, out);
}